// ARMA_15453292331025
// MI455X (gfx1250) — compile-verified
//
#include <hip/hip_runtime.h>

// ---------------- problem constants (match reference) ----------------
#define NN   50000
#define EE   600000
#define FIN  128
#define HIDD 128
#define CLSS 8
#define KS   3
#define GG   64

typedef float v2f __attribute__((ext_vector_type(2)));
typedef float v8f __attribute__((ext_vector_type(8)));

// ---------------- utility kernels ----------------
__global__ __launch_bounds__(256) void zero_f32(float* p, long long n) {
  long long i = (long long)blockIdx.x * blockDim.x + threadIdx.x;
  if (i < n) p[i] = 0.f;
}

__global__ __launch_bounds__(256) void deg_kernel(const int* __restrict__ col,
                                                  float* __restrict__ deg, int E) {
  int e = blockIdx.x * blockDim.x + threadIdx.x;
  if (e < E) atomicAdd(&deg[col[e]], 1.0f);
}

__global__ __launch_bounds__(256) void dis_kernel(float* deg, int n) {
  int i = blockIdx.x * blockDim.x + threadIdx.x;
  if (i < n) {
    float d = deg[i];
    deg[i] = d > 0.f ? rsqrtf(fmaxf(d, 1.f)) : 0.f;
  }
}

__global__ __launch_bounds__(256) void norm_kernel(const int* __restrict__ row,
                                                   const int* __restrict__ col,
                                                   const float* __restrict__ dis,
                                                   float* __restrict__ norm, int E) {
  int e = blockIdx.x * blockDim.x + threadIdx.x;
  if (e < E) norm[e] = dis[row[e]] * dis[col[e]];
}

// W (K,F,H) -> Wt (K,H,F) so B fragments are contiguous b64 loads
__global__ __launch_bounds__(256) void transpose_w(const float* __restrict__ W,
                                                   float* __restrict__ Wt,
                                                   int K, int F, int H) {
  int idx = blockIdx.x * blockDim.x + threadIdx.x;
  int total = K * F * H;
  if (idx >= total) return;
  int h = idx % H;
  int t = idx / H;
  int f = t % F;
  int k = t / F;
  Wt[((size_t)k * H + h) * F + f] = W[idx];
}

// ---------------- fp32 WMMA GEMM: Y[k] = X(sel) x W[k] ----------------
// X: (N,F) row-major (or (K,N,F) if x_per_k), Wt: (K,H,F) transposed, Y: (K,N,H)
// Requires Nrows%16==0, F%4==0, H multiple of 64*blockDim.y (all exact here).
// Each wave computes a 16x64 strip; fragments are double-buffered so the
// next K-step's 5 b64 loads are in flight while the current 4 WMMAs execute.
__global__ __launch_bounds__(64) void wmma_gemm_f32(
    const float* __restrict__ X, const float* __restrict__ Wt,
    float* __restrict__ Y, int Nrows, int F, int H, int x_per_k) {
  const int lane = threadIdx.x;                     // 0..31, one wave per y-slice
  const int tm   = blockIdx.x;                      // M tile (16 rows)
  const int k    = blockIdx.z;
  const int hbase = (blockIdx.y * blockDim.y + threadIdx.y) * 64;

  const float* Xk = X + (x_per_k ? (size_t)k * Nrows * F : (size_t)0);
  const float* Wk = Wt + (size_t)k * H * F;
  float*       Yk = Y + (size_t)k * Nrows * H;

  const int mrow = tm * 16 + (lane & 15);           // A row for this lane
  const int hcol = lane & 15;                       // column within each 16-wide tile
  const int kb   = (lane >> 4) * 2;                 // K sub-offset: 0 or 2

  // Contiguous {K=kb, K=kb+1} pairs for both A and B -> b64 loads
  const float* ap = Xk + (size_t)mrow * F + kb;
  const float* bp = Wk + (size_t)(hbase + hcol) * F + kb;
  const size_t bs = (size_t)16 * F;                 // stride between the 4 B tiles

  v8f acc0 = {}, acc1 = {}, acc2 = {}, acc3 = {};

  // ---- prologue: load fragments for K-step 0 ----
  v2f a, b0, b1, b2, b3;
  a.x  = ap[0];        a.y  = ap[1];
  b0.x = bp[0];        b0.y = bp[1];
  b1.x = bp[bs];       b1.y = bp[bs + 1];
  b2.x = bp[2 * bs];   b2.y = bp[2 * bs + 1];
  b3.x = bp[3 * bs];   b3.y = bp[3 * bs + 1];

#pragma unroll 2
  for (int f0 = 0; f0 < F - 4; f0 += 4) {
    // issue next K-step's loads before consuming current fragments
    ap += 4;
    bp += 4;
    v2f an, b0n, b1n, b2n, b3n;
    an.x  = ap[0];        an.y  = ap[1];
    b0n.x = bp[0];        b0n.y = bp[1];
    b1n.x = bp[bs];       b1n.y = bp[bs + 1];
    b2n.x = bp[2 * bs];   b2n.y = bp[2 * bs + 1];
    b3n.x = bp[3 * bs];   b3n.y = bp[3 * bs + 1];

    acc0 = __builtin_amdgcn_wmma_f32_16x16x4_f32(false, a, false, b0, (short)0, acc0, false, false);
    acc1 = __builtin_amdgcn_wmma_f32_16x16x4_f32(false, a, false, b1, (short)0, acc1, false, false);
    acc2 = __builtin_amdgcn_wmma_f32_16x16x4_f32(false, a, false, b2, (short)0, acc2, false, false);
    acc3 = __builtin_amdgcn_wmma_f32_16x16x4_f32(false, a, false, b3, (short)0, acc3, false, false);

    a = an; b0 = b0n; b1 = b1n; b2 = b2n; b3 = b3n;
  }

  // ---- epilogue: last K-step ----
  acc0 = __builtin_amdgcn_wmma_f32_16x16x4_f32(false, a, false, b0, (short)0, acc0, false, false);
  acc1 = __builtin_amdgcn_wmma_f32_16x16x4_f32(false, a, false, b1, (short)0, acc1, false, false);
  acc2 = __builtin_amdgcn_wmma_f32_16x16x4_f32(false, a, false, b2, (short)0, acc2, false, false);
  acc3 = __builtin_amdgcn_wmma_f32_16x16x4_f32(false, a, false, b3, (short)0, acc3, false, false);

  // C/D 16x16 fp32: vgpr i -> M=i (lanes 0-15) / M=i+8 (lanes 16-31)
  const int rbase = tm * 16 + ((lane >> 4) ? 8 : 0);
  float* yp = Yk + (size_t)rbase * H + hbase + hcol;
#pragma unroll
  for (int i = 0; i < 8; ++i) {
    yp[(size_t)i * H +  0] = acc0[i];
    yp[(size_t)i * H + 16] = acc1[i];
    yp[(size_t)i * H + 32] = acc2[i];
    yp[(size_t)i * H + 48] = acc3[i];
  }
}

// ---------------- small VALU GEMM for the thin layer-2 matmuls ----------------
__global__ __launch_bounds__(256) void small_gemm(
    const float* __restrict__ X, const float* __restrict__ W,
    float* __restrict__ Y, int Nrows, int F, int C, int K, int x_per_k) {
  long long idx = (long long)blockIdx.x * blockDim.x + threadIdx.x;
  long long total = (long long)K * Nrows * C;
  if (idx >= total) return;
  int c = (int)(idx % C);
  long long t = idx / C;
  int n = (int)(t % Nrows);
  int k = (int)(t / Nrows);
  const float* Xr = X + (x_per_k ? ((size_t)k * Nrows + n) * F : (size_t)n * F);
  const float* Wk = W + (size_t)k * F * C;
  float s = 0.f;
  for (int f = 0; f < F; ++f) s += Xr[f] * Wk[(size_t)f * C + c];
  Y[idx] = s;  // (k,n,c) contiguous
}

// ---------------- edge-parallel gather*norm -> atomic scatter (float4) -------
// Requires H % 4 == 0 (128 and 8 both qualify).
__global__ __launch_bounds__(256) void propagate4(
    const float* __restrict__ h, const int* __restrict__ row,
    const int* __restrict__ col, const float* __restrict__ norm,
    float* __restrict__ agg, int K, int E, int H, int Nrows) {
  const int Hq = H >> 2;
  long long idx = (long long)blockIdx.x * blockDim.x + threadIdx.x;
  long long total = (long long)K * E * Hq;
  if (idx >= total) return;
  int f4 = (int)(idx % Hq);
  long long t = idx / Hq;
  int e = (int)(t % E);
  int k = (int)(t / E);
  int r = row[e];
  int c = col[e];
  float nm = norm[e];
  const float4 hv = *(const float4*)(h + ((size_t)k * Nrows + r) * H + 4 * f4);
  float* dst = agg + ((size_t)k * Nrows + c) * H + 4 * f4;
  atomicAdd(dst + 0, hv.x * nm);
  atomicAdd(dst + 1, hv.y * nm);
  atomicAdd(dst + 2, hv.z * nm);
  atomicAdd(dst + 3, hv.w * nm);
}

// ---------------- out = act(agg + root + bias) ----------------
__global__ __launch_bounds__(256) void bias_act(
    const float* __restrict__ agg, const float* __restrict__ root,
    const float* __restrict__ bias, float* __restrict__ out,
    int K, int Nrows, int H, int do_relu) {
  long long idx = (long long)blockIdx.x * blockDim.x + threadIdx.x;
  long long total = (long long)K * Nrows * H;
  if (idx >= total) return;
  int f = (int)(idx % H);
  int k = (int)(idx / ((long long)Nrows * H));
  float v = agg[idx] + root[idx] + bias[(size_t)k * H + f];
  out[idx] = do_relu ? fmaxf(v, 0.f) : v;
}

// ---------------- mean over K stacks (optional relu) ----------------
__global__ __launch_bounds__(256) void mean_k(
    const float* __restrict__ in, float* __restrict__ out,
    int K, int Nrows, int H, int do_relu) {
  long long idx = (long long)blockIdx.x * blockDim.x + threadIdx.x;
  long long total = (long long)Nrows * H;
  if (idx >= total) return;
  float s = 0.f;
  for (int k = 0; k < K; ++k) s += in[(size_t)k * Nrows * H + idx];
  s *= (1.0f / (float)K);
  out[idx] = do_relu ? fmaxf(s, 0.f) : s;
}

// ---------------- global_add_pool via atomics ----------------
__global__ __launch_bounds__(256) void pool_kernel(
    const float* __restrict__ val, const int* __restrict__ batch,
    float* __restrict__ dst, int Nrows, int H) {
  long long idx = (long long)blockIdx.x * blockDim.x + threadIdx.x;
  long long total = (long long)Nrows * H;
  if (idx >= total) return;
  int n = (int)(idx / H);
  int f = (int)(idx % H);
  atomicAdd(&dst[(size_t)batch[n] * H + f], val[idx]);
}

// ---------------- host-side orchestration ----------------
static inline unsigned nblk(long long n, int b) { return (unsigned)((n + b - 1) / b); }

extern "C" void kernel_launch(void* const* d_in, const int* in_sizes, int n_in,
                              void* d_out, int out_size, void* d_ws, size_t ws_size,
                              hipStream_t stream) {
  const float* x         = (const float*)d_in[0];
  const int*   ei        = (const int*)  d_in[1];
  const int*   batch     = (const int*)  d_in[2];
  const float* c1_init_w = (const float*)d_in[3];
  const float* c1_w      = (const float*)d_in[4];
  const float* c1_root_w = (const float*)d_in[5];
  const float* c1_bias   = (const float*)d_in[6];
  const float* c2_init_w = (const float*)d_in[7];
  const float* c2_w      = (const float*)d_in[8];
  const float* c2_root_w = (const float*)d_in[9];
  const float* c2_bias   = (const float*)d_in[10];
  float* out = (float*)d_out;

  const int* row = ei;        // edge_index[0,:]
  const int* col = ei + EE;   // edge_index[1,:]

  // ---- workspace carve-up (floats) ----
  float* ws   = (float*)d_ws;
  float* norm = ws;  ws += EE;
  float* dis  = ws;  ws += NN;
  float* wt   = ws;  ws += (size_t)KS * FIN * HIDD;   // transposed weights (49152)
  const size_t big = (size_t)KS * NN * HIDD;
  float* root = ws;  ws += big;   // layer-1 root term (persists over T)
  float* bufA = ws;  ws += big;   // GEMM output / messages source
  float* bufB = ws;  ws += big;   // aggregation (atomic scatter target)
  float* bufC = ws;  ws += big;   // activated output
  float* hbuf = ws;  ws += (size_t)NN * HIDD;  // layer-1 result / layer-2 mean

  // ---- gcn_norm ----
  zero_f32<<<nblk(NN, 256), 256, 0, stream>>>(dis, NN);
  deg_kernel<<<nblk(EE, 256), 256, 0, stream>>>(col, dis, EE);
  dis_kernel<<<nblk(NN, 256), 256, 0, stream>>>(dis, NN);
  norm_kernel<<<nblk(EE, 256), 256, 0, stream>>>(row, col, dis, norm, EE);

  // ---- layer 1 (ARMA, relu) ----
  dim3 wb(32, 2);                 // 2 waves, each owns a 16x64 output strip
  dim3 wg1(NN / 16, 1, KS);       // 3125 x (H/128=1) x 3, all exact
  const int wtn = KS * FIN * HIDD;
  const long long aggN  = (long long)KS * NN * HIDD;
  const long long propN = (long long)KS * EE * (HIDD / 4);

  transpose_w<<<nblk(wtn, 256), 256, 0, stream>>>(c1_root_w, wt, KS, FIN, HIDD);
  wmma_gemm_f32<<<wg1, wb, 0, stream>>>(x, wt, root, NN, FIN, HIDD, 0);

  // t = 0
  transpose_w<<<nblk(wtn, 256), 256, 0, stream>>>(c1_init_w, wt, KS, FIN, HIDD);
  wmma_gemm_f32<<<wg1, wb, 0, stream>>>(x, wt, bufA, NN, FIN, HIDD, 0);
  zero_f32<<<nblk(aggN, 256), 256, 0, stream>>>(bufB, aggN);
  propagate4<<<nblk(propN, 256), 256, 0, stream>>>(bufA, row, col, norm, bufB, KS, EE, HIDD, NN);
  bias_act<<<nblk(aggN, 256), 256, 0, stream>>>(bufB, root, c1_bias, bufC, KS, NN, HIDD, 1);

  // t = 1
  transpose_w<<<nblk(wtn, 256), 256, 0, stream>>>(c1_w, wt, KS, HIDD, HIDD);
  wmma_gemm_f32<<<wg1, wb, 0, stream>>>(bufC, wt, bufA, NN, HIDD, HIDD, 1);
  zero_f32<<<nblk(aggN, 256), 256, 0, stream>>>(bufB, aggN);
  propagate4<<<nblk(propN, 256), 256, 0, stream>>>(bufA, row, col, norm, bufB, KS, EE, HIDD, NN);
  bias_act<<<nblk(aggN, 256), 256, 0, stream>>>(bufB, root, c1_bias, bufC, KS, NN, HIDD, 1);

  // h = relu(mean_k)
  const long long nhN = (long long)NN * HIDD;
  mean_k<<<nblk(nhN, 256), 256, 0, stream>>>(bufC, hbuf, KS, NN, HIDD, 1);

  // ---- outputs: zero then pool features into d_out[G*CLS ..) ----
  const long long outN = (long long)GG * CLSS + (long long)GG * HIDD;
  zero_f32<<<nblk(outN, 256), 256, 0, stream>>>(out, outN);
  pool_kernel<<<nblk(nhN, 256), 256, 0, stream>>>(hbuf, batch, out + GG * CLSS, NN, HIDD);

  // ---- layer 2 (ARMA, identity act), H=8 — reuse big buffers ----
  const long long n2 = (long long)KS * NN * CLSS;
  const long long p2 = (long long)KS * EE * (CLSS / 4);
  float* root2 = root;
  float* A2 = bufA;
  float* B2 = bufB;
  float* C2 = bufC;

  small_gemm<<<nblk(n2, 256), 256, 0, stream>>>(hbuf, c2_root_w, root2, NN, HIDD, CLSS, KS, 0);

  // t = 0
  small_gemm<<<nblk(n2, 256), 256, 0, stream>>>(hbuf, c2_init_w, A2, NN, HIDD, CLSS, KS, 0);
  zero_f32<<<nblk(n2, 256), 256, 0, stream>>>(B2, n2);
  propagate4<<<nblk(p2, 256), 256, 0, stream>>>(A2, row, col, norm, B2, KS, EE, CLSS, NN);
  bias_act<<<nblk(n2, 256), 256, 0, stream>>>(B2, root2, c2_bias, C2, KS, NN, CLSS, 0);

  // t = 1
  small_gemm<<<nblk(n2, 256), 256, 0, stream>>>(C2, c2_w, A2, NN, CLSS, CLSS, KS, 1);
  zero_f32<<<nblk(n2, 256), 256, 0, stream>>>(B2, n2);
  propagate4<<<nblk(p2, 256), 256, 0, stream>>>(A2, row, col, norm, B2, KS, EE, CLSS, NN);
  bias_act<<<nblk(n2, 256), 256, 0, stream>>>(B2, root2, c2_bias, C2, KS, NN, CLSS, 0);

  // mean over K (no relu), then pool into d_out[0 .. G*CLS)
  const long long ncN = (long long)NN * CLSS;
  mean_k<<<nblk(ncN, 256), 256, 0, stream>>>(C2, hbuf, KS, NN, CLSS, 0);
  pool_kernel<<<nblk(ncN, 256), 256, 0, stream>>>(hbuf, batch, out, NN, CLSS);
}